// DTIGraph3EdgePoolLayer_68745246539847
// MI455X (gfx1250) — compile-verified
//
#include <hip/hip_runtime.h>
#include <math.h>

typedef float v2f __attribute__((ext_vector_type(2)));
typedef float v8f __attribute__((ext_vector_type(8)));

#define DIMF 128
#define LDA 132   // 16-row A-tile stride (pad: 132%64=4 -> conflict-free frag reads)
#define LDG 772   // gate buffer stride (772%64=4)

static __device__ __forceinline__ float lrelu(float x) {
    return x > 0.0f ? x : 0.01f * x;
}

static __device__ __forceinline__ v8f wmma_f32x4(v2f a, v2f b, v8f c) {
    // D = A(16x4, f32) * B(4x16, f32) + C(16x16, f32)
    return __builtin_amdgcn_wmma_f32_16x16x4_f32(false, a, false, b, (short)0, c, false, false);
}

// ---------------------------------------------------------------------------
// K1: gdot[g] = dot(lrelu(g_feats[g]), logits_w[0:128])   (one wave per graph)
// ---------------------------------------------------------------------------
__global__ __launch_bounds__(256) void k_gdot(const float* __restrict__ g_feats,
                                              const float* __restrict__ logits_w,
                                              float* __restrict__ gdot, int G) {
    int w = (int)((blockIdx.x * blockDim.x + threadIdx.x) >> 5);
    int lane = threadIdx.x & 31;
    if (w >= G) return;
    float4 v = *(const float4*)(g_feats + (size_t)w * DIMF + lane * 4);
    float4 ww = *(const float4*)(logits_w + lane * 4);
    float s = lrelu(v.x) * ww.x + lrelu(v.y) * ww.y + lrelu(v.z) * ww.z + lrelu(v.w) * ww.w;
    for (int d = 16; d; d >>= 1) s += __shfl_xor(s, d, 32);
    if (lane == 0) gdot[w] = s;
}

// ---------------------------------------------------------------------------
// K2: z[e] = lrelu(gdot[seg[e]] + dot(edge_feats[e], logits_w[128:256]) + b)
// one wave per edge, fully coalesced 512B row read
// ---------------------------------------------------------------------------
__global__ __launch_bounds__(256) void k_z(const float* __restrict__ edge_feats,
                                           const float* __restrict__ logits_w,
                                           const float* __restrict__ logits_b,
                                           const int* __restrict__ seg,
                                           const float* __restrict__ gdot,
                                           float* __restrict__ z_ws, int E) {
    int e = (int)((blockIdx.x * blockDim.x + threadIdx.x) >> 5);
    int lane = threadIdx.x & 31;
    if (e >= E) return;
    float4 v = *(const float4*)(edge_feats + (size_t)e * DIMF + lane * 4);
    float4 ww = *(const float4*)(logits_w + DIMF + lane * 4);
    float s = v.x * ww.x + v.y * ww.y + v.z * ww.z + v.w * ww.w;
    for (int d = 16; d; d >>= 1) s += __shfl_xor(s, d, 32);
    if (lane == 0) {
        z_ws[e] = lrelu(s + gdot[seg[e]] + logits_b[0]);
    }
}

// ---------------------------------------------------------------------------
// K3: CSR offsets via binary search (seg is sorted)
// ---------------------------------------------------------------------------
__global__ __launch_bounds__(256) void k_off(const int* __restrict__ seg,
                                             int* __restrict__ off, int E, int G) {
    int g = blockIdx.x * blockDim.x + threadIdx.x;
    if (g > G) return;
    if (g == G) { off[G] = E; return; }
    int lo = 0, hi = E;
    while (lo < hi) {
        int mid = (lo + hi) >> 1;
        if (seg[mid] < g) lo = mid + 1; else hi = mid;
    }
    off[g] = lo;
}

// ---------------------------------------------------------------------------
// K4: per-graph block: softmax (max/denom/a) + WMMA projection GEMM with fused
// lrelu + ez-weighted segment reduction -> context[g]
// ---------------------------------------------------------------------------
__global__ __launch_bounds__(256) void k_main(const float* __restrict__ edge_feats,
                                              const float* __restrict__ proj_w,
                                              const float* __restrict__ proj_b,
                                              const float* __restrict__ z_ws,
                                              const int* __restrict__ off,
                                              float* __restrict__ a_out,
                                              float* __restrict__ ctx_ws) {
    __shared__ float sA[16 * LDA];
    __shared__ float sEz[16];
    __shared__ float sRed[8];
    __shared__ float sZmax;
    __shared__ float sInvDen;

    const int g = blockIdx.x;
    const int t = threadIdx.x;
    const int wave = t >> 5;
    const int lane = t & 31;
    const int lo = lane & 15;
    const int hi = lane >> 4;

    const int s = off[g];
    const int cnt = off[g + 1] - s;

    // --- segment max ---
    float lm = -3.402823466e38f;
    for (int i = t; i < cnt; i += 256) lm = fmaxf(lm, z_ws[s + i]);
    for (int d = 16; d; d >>= 1) lm = fmaxf(lm, __shfl_xor(lm, d, 32));
    if (lane == 0) sRed[wave] = lm;
    __syncthreads();
    if (t == 0) {
        float m = sRed[0];
        for (int i = 1; i < 8; i++) m = fmaxf(m, sRed[i]);
        sZmax = m;
    }
    __syncthreads();
    const float zmax = sZmax;

    // --- segment sum of exp ---
    float ls = 0.0f;
    for (int i = t; i < cnt; i += 256) ls += expf(z_ws[s + i] - zmax);
    for (int d = 16; d; d >>= 1) ls += __shfl_xor(ls, d, 32);
    if (lane == 0) sRed[wave] = ls;
    __syncthreads();
    if (t == 0) {
        float den = 0.0f;
        for (int i = 0; i < 8; i++) den += sRed[i];
        sInvDen = 1.0f / fmaxf(den, 1e-20f);
    }
    __syncthreads();
    const float invden = sInvDen;

    // --- attention output a[e] ---
    for (int i = t; i < cnt; i += 256)
        a_out[s + i] = expf(z_ws[s + i] - zmax) * invden;

    // --- WMMA projection + weighted accumulation ---
    // wave owns features [n0, n0+16); preload all B fragments of proj_w
    const int n0 = wave * 16;
    v2f breg[32];
    {
        const float* wp = proj_w + (size_t)(n0 + lo) * DIMF;
        #pragma unroll
        for (int kk = 0; kk < 32; kk++) {
            int k = kk * 4 + 2 * hi;
            breg[kk] = *(const v2f*)(wp + k);   // B[k][n]=proj_w[n][k], consecutive pair
        }
    }
    const float pb = proj_b[n0 + lo];

    float acc = 0.0f;
    const int ntiles = (cnt + 15) >> 4;
    const int r = t >> 4;           // staging row 0..15
    const int cb = (t & 15) * 8;    // staging col base

    for (int tile = 0; tile < ntiles; tile++) {
        __syncthreads();
        // stage 16x128 A tile (zero-padded) + ez values
        {
            int er = tile * 16 + r;
            float4 v0 = make_float4(0.f, 0.f, 0.f, 0.f);
            float4 v1 = v0;
            if (er < cnt) {
                const float4* src = (const float4*)(edge_feats + (size_t)(s + er) * DIMF + cb);
                v0 = src[0];
                v1 = src[1];
            }
            *(float4*)&sA[r * LDA + cb] = v0;
            *(float4*)&sA[r * LDA + cb + 4] = v1;
            if (t < 16) {
                int em = tile * 16 + t;
                sEz[t] = (em < cnt) ? expf(z_ws[s + em] - zmax) : 0.0f;
            }
            // prefetch next tile's rows to overlap HBM latency with WMMA compute
            if (er + 16 < cnt) {
                __builtin_prefetch(edge_feats + (size_t)(s + er + 16) * DIMF + cb, 0, 0);
            }
        }
        __syncthreads();

        // two independent WMMA accumulation chains (even/odd K-steps) for ILP
        v8f c0 = {0.f, 0.f, 0.f, 0.f, 0.f, 0.f, 0.f, 0.f};
        v8f c1 = {0.f, 0.f, 0.f, 0.f, 0.f, 0.f, 0.f, 0.f};
        #pragma unroll
        for (int kk = 0; kk < 32; kk += 2) {
            int k0 = kk * 4 + 2 * hi;
            int k1 = (kk + 1) * 4 + 2 * hi;
            v2f a0 = *(const v2f*)&sA[lo * LDA + k0];   // A: M=lo, K=k0..k0+1
            v2f a1 = *(const v2f*)&sA[lo * LDA + k1];
            c0 = wmma_f32x4(a0, breg[kk], c0);
            c1 = wmma_f32x4(a1, breg[kk + 1], c1);
        }
        // C frag: lane (lo,hi), VGPR j -> element (m = j+8*hi, n = n0+lo)
        #pragma unroll
        for (int j = 0; j < 8; j++) {
            float hv = lrelu(c0[j] + c1[j] + pb);
            acc += sEz[j + 8 * hi] * hv;
        }
    }

    // combine the two M-halves; both halves then hold the full sum for n0+lo
    acc += __shfl_xor(acc, 16, 32);
    if (lane < 16) {
        float ctx = lrelu(acc * invden);
        ctx_ws[(size_t)g * DIMF + n0 + lo] = ctx;
    }
}

// ---------------------------------------------------------------------------
// K5: GRU cell for 16 graphs per block: GI = ctx@wih^T, GH = g_feats@whh^T via
// WMMA, then fused gate math; out = lrelu(h)
// ---------------------------------------------------------------------------
__global__ __launch_bounds__(256) void k_gru(const float* __restrict__ ctx_ws,
                                             const float* __restrict__ g_feats,
                                             const float* __restrict__ wih,
                                             const float* __restrict__ whh,
                                             const float* __restrict__ bih,
                                             const float* __restrict__ bhh,
                                             float* __restrict__ out, int G) {
    __shared__ float sC[16 * LDA];
    __shared__ float sG[16 * LDA];
    __shared__ float sR[16 * LDG];   // cols 0..383 = GI, 384..767 = GH

    const int g0 = blockIdx.x * 16;
    const int t = threadIdx.x;
    const int wave = t >> 5;
    const int lane = t & 31;
    const int lo = lane & 15;
    const int hi = lane >> 4;

    // stage 16 rows of context and g_feats
    {
        int r = t >> 4;
        int cb = (t & 15) * 8;
        float4 c0 = make_float4(0.f, 0.f, 0.f, 0.f), c1 = c0, g0v = c0, g1v = c0;
        if (g0 + r < G) {
            const float4* cs = (const float4*)(ctx_ws + (size_t)(g0 + r) * DIMF + cb);
            const float4* gs = (const float4*)(g_feats + (size_t)(g0 + r) * DIMF + cb);
            c0 = cs[0]; c1 = cs[1];
            g0v = gs[0]; g1v = gs[1];
        }
        *(float4*)&sC[r * LDA + cb] = c0;
        *(float4*)&sC[r * LDA + cb + 4] = c1;
        *(float4*)&sG[r * LDA + cb] = g0v;
        *(float4*)&sG[r * LDA + cb + 4] = g1v;
    }
    __syncthreads();

    // 48 N-tiles: 24 for GI (wih / context), 24 for GH (whh / g_feats)
    for (int nt = wave; nt < 48; nt += 8) {
        const bool isI = nt < 24;
        const int n0 = (isI ? nt : nt - 24) * 16;
        const float* W = isI ? wih : whh;
        const float* At = isI ? sC : sG;
        const int dst = (isI ? 0 : 384) + n0;
        const float* wp = W + (size_t)(n0 + lo) * DIMF;

        v8f c0 = {0.f, 0.f, 0.f, 0.f, 0.f, 0.f, 0.f, 0.f};
        v8f c1 = {0.f, 0.f, 0.f, 0.f, 0.f, 0.f, 0.f, 0.f};
        #pragma unroll
        for (int kk = 0; kk < 32; kk += 2) {
            int k0 = kk * 4 + 2 * hi;
            int k1 = (kk + 1) * 4 + 2 * hi;
            v2f a0 = *(const v2f*)&At[lo * LDA + k0];
            v2f a1 = *(const v2f*)&At[lo * LDA + k1];
            v2f b0 = *(const v2f*)(wp + k0);
            v2f b1 = *(const v2f*)(wp + k1);
            c0 = wmma_f32x4(a0, b0, c0);
            c1 = wmma_f32x4(a1, b1, c1);
        }
        #pragma unroll
        for (int j = 0; j < 8; j++) {
            sR[(j + 8 * hi) * LDG + dst + lo] = c0[j] + c1[j];
        }
    }
    __syncthreads();

    // fused gate math over 16x128 outputs
    for (int idx = t; idx < 16 * DIMF; idx += 256) {
        int m = idx >> 7;
        int d = idx & 127;
        if (g0 + m >= G) continue;
        float ir  = sR[m * LDG + d]       + bih[d];
        float iz  = sR[m * LDG + 128 + d] + bih[128 + d];
        float inn = sR[m * LDG + 256 + d] + bih[256 + d];
        float hr  = sR[m * LDG + 384 + d] + bhh[d];
        float hz  = sR[m * LDG + 512 + d] + bhh[128 + d];
        float hn  = sR[m * LDG + 640 + d] + bhh[256 + d];
        float rg = 1.0f / (1.0f + expf(-(ir + hr)));
        float zg = 1.0f / (1.0f + expf(-(iz + hz)));
        float n  = tanhf(inn + rg * hn);
        float gf = sG[m * LDA + d];
        float h  = (1.0f - zg) * n + zg * gf;
        out[(size_t)(g0 + m) * DIMF + d] = lrelu(h);
    }
}

// ---------------------------------------------------------------------------
extern "C" void kernel_launch(void* const* d_in, const int* in_sizes, int n_in,
                              void* d_out, int out_size, void* d_ws, size_t ws_size,
                              hipStream_t stream) {
    const float* edge_feats = (const float*)d_in[0];
    const float* g_feats    = (const float*)d_in[1];
    const int*   seg        = (const int*)d_in[2];
    const float* logits_w   = (const float*)d_in[3];
    const float* logits_b   = (const float*)d_in[4];
    const float* proj_w     = (const float*)d_in[5];
    const float* proj_b     = (const float*)d_in[6];
    const float* gru_wih    = (const float*)d_in[7];
    const float* gru_whh    = (const float*)d_in[8];
    const float* gru_bih    = (const float*)d_in[9];
    const float* gru_bhh    = (const float*)d_in[10];

    const int E = in_sizes[2];
    const int G = in_sizes[1] / DIMF;

    float* ws     = (float*)d_ws;
    float* gdot   = ws;                                   // G
    float* z_ws   = ws + G;                               // E
    float* ctx_ws = ws + G + E;                           // G*128
    int*   off    = (int*)(ws + (size_t)G + E + (size_t)G * DIMF); // G+1

    float* h_out = (float*)d_out;              // G*128
    float* a_out = h_out + (size_t)G * DIMF;   // E

    k_gdot<<<(G + 7) / 8, 256, 0, stream>>>(g_feats, logits_w, gdot, G);
    k_z<<<(E + 7) / 8, 256, 0, stream>>>(edge_feats, logits_w, logits_b, seg, gdot, z_ws, E);
    k_off<<<(G + 1 + 255) / 256, 256, 0, stream>>>(seg, off, E, G);
    k_main<<<G, 256, 0, stream>>>(edge_feats, proj_w, proj_b, z_ws, off, a_out, ctx_ws);
    k_gru<<<(G + 15) / 16, 256, 0, stream>>>(ctx_ws, g_feats, gru_wih, gru_whh, gru_bih, gru_bhh, h_out, G);
}